// Conjugate_Gradient_40785009443616
// MI455X (gfx1250) — compile-verified
//
#include <hip/hip_runtime.h>
#include <hip/hip_bf16.h>

// CG solve, 8 batches of 4096x4096 SPD systems, 20 iterations.
// Memory-bound: 512MB of M streamed per iteration. Matvec done with
// V_WMMA_F32_16X16X4_F32 exploiting symmetry of M (y = M^T p = M p).

#define CG_N     4096
#define CG_B     8
#define CG_KC    8                  // K-dimension split for parallelism
#define CG_KCH   (CG_N / CG_KC)     // 512 k's per chunk
#define CG_NITER 20
#define CG_TOL   1e-10f

typedef __attribute__((ext_vector_type(2))) float v2f;
typedef __attribute__((ext_vector_type(8))) float v8f;

// ---------------------------------------------------------------------------
// Wave32 reduction (CDNA5 is wave32-only)
// ---------------------------------------------------------------------------
__device__ __forceinline__ float wave_reduce_sum(float v) {
    #pragma unroll
    for (int off = 16; off > 0; off >>= 1)
        v += __shfl_xor(v, off, 32);
    return v;
}

// ---------------------------------------------------------------------------
// WMMA matvec: parts[kc][b][n] = sum_{k in chunk kc} M[b][k][n] * vec[b][k]
// (M symmetric => this equals the row-major matvec result.)
// A-matrix (16x4 f32): p-values replicated across all 16 M-rows.
//   VGPR0: lanes 0-15 K=0, lanes 16-31 K=2 ; VGPR1: K=1 / K=3.
// B-matrix (4x16 f32): tile of M, coalesced along n (64B per half-wave).
// D: every row identical; row 0 (VGPR0, lanes 0-15) = 16 outputs.
// ---------------------------------------------------------------------------
__global__ __launch_bounds__(32)
void cg_matvec_wmma(const float* __restrict__ M,
                    const float* __restrict__ vec,
                    float* __restrict__ parts) {
    const int lane = threadIdx.x;        // 0..31, one wave per block
    const int n0   = blockIdx.x * 32;    // 32 output columns per wave
    const int kc   = blockIdx.y;         // K chunk
    const int b    = blockIdx.z;         // batch

    const float* Mb = M + (size_t)b * CG_N * CG_N;
    const float* vb = vec + b * CG_N;

    const int hi  = lane >> 4;           // 0: lanes 0-15, 1: lanes 16-31
    const int col = lane & 15;

    v8f acc0 = {};
    v8f acc1 = {};

    int k0 = kc * CG_KCH;
    const int kend = k0 + CG_KCH;
    for (; k0 < kend; k0 += 4) {
        // A: replicated vector values per documented 16x4 f32 layout
        const float* pa = vb + k0 + (hi << 1);
        v2f a;
        a.x = pa[0];                     // K=0 (lo) / K=2 (hi)
        a.y = pa[1];                     // K=1 (lo) / K=3 (hi)

        // B: rows k0+2*hi and k0+2*hi+1 of M, columns n0..n0+31
        const int   row = k0 + (hi << 1);
        const float* m0 = Mb + (size_t)row * CG_N + n0;
        const float* m1 = m0 + CG_N;

        // prefetch the M stream a few rows ahead (global_prefetch_b8)
        __builtin_prefetch(m0 + (size_t)16 * CG_N, 0, 0);

        v2f b0, b1;
        b0.x = m0[col];        b0.y = m1[col];
        b1.x = m0[col + 16];   b1.y = m1[col + 16];

        acc0 = __builtin_amdgcn_wmma_f32_16x16x4_f32(
                   false, a, false, b0, (short)0, acc0, false, false);
        acc1 = __builtin_amdgcn_wmma_f32_16x16x4_f32(
                   false, a, false, b1, (short)0, acc1, false, false);
    }

    // Row m=0 of D: VGPR0, lanes 0-15 hold y[n0+lane] (all rows identical)
    if (lane < 16) {
        float* dst = parts + ((size_t)(kc * CG_B + b)) * CG_N + n0;
        dst[col]      = acc0[0];
        dst[col + 16] = acc1[0];
    }
}

// ---------------------------------------------------------------------------
// Zero the scalar slots (rTr[NITER+1] followed by pAp[NITER], contiguous)
// ---------------------------------------------------------------------------
__global__ void cg_init_scalars(float* __restrict__ scal, int n) {
    int i = threadIdx.x;
    if (i < n) scal[i] = 0.0f;
}

// ---------------------------------------------------------------------------
// r0 = RHS - M X ; p0 = r0 ; Xout = Xin ; rTr[0] += sum(r0^2)
// ---------------------------------------------------------------------------
__global__ __launch_bounds__(256)
void cg_init_rp(const float* __restrict__ RHS, const float* __restrict__ Xin,
                const float* __restrict__ parts,
                float* __restrict__ r, float* __restrict__ p,
                float* __restrict__ Xout, float* __restrict__ rTr0) {
    const int e = blockIdx.x * blockDim.x + threadIdx.x;   // 0..B*N-1
    float ap = 0.0f;
    #pragma unroll
    for (int c = 0; c < CG_KC; ++c)
        ap += parts[(size_t)c * (CG_B * CG_N) + e];
    const float r0 = RHS[e] - ap;
    r[e] = r0;
    p[e] = r0;
    Xout[e] = Xin[e];
    const float s = wave_reduce_sum(r0 * r0);
    if ((threadIdx.x & 31) == 0) atomicAdd(rTr0, s);
}

// ---------------------------------------------------------------------------
// Ap = sum of K-chunk partials ; pAp[i] += sum(p * Ap)
// ---------------------------------------------------------------------------
__global__ __launch_bounds__(256)
void cg_apdot(const float* __restrict__ parts, const float* __restrict__ p,
              float* __restrict__ Ap, float* __restrict__ pApSlot) {
    const int e = blockIdx.x * blockDim.x + threadIdx.x;
    float ap = 0.0f;
    #pragma unroll
    for (int c = 0; c < CG_KC; ++c)
        ap += parts[(size_t)c * (CG_B * CG_N) + e];
    Ap[e] = ap;
    const float s = wave_reduce_sum(p[e] * ap);
    if ((threadIdx.x & 31) == 0) atomicAdd(pApSlot, s);
}

// ---------------------------------------------------------------------------
// alpha = rTr/pAp (if active) ; X += alpha p ; r -= alpha Ap ;
// rTrNext += sum(r_new^2)   (active gate is wave-uniform)
// ---------------------------------------------------------------------------
__global__ __launch_bounds__(256)
void cg_update1(const float* __restrict__ rTrI, const float* __restrict__ pApI,
                const float* __restrict__ p, const float* __restrict__ Ap,
                float* __restrict__ X, float* __restrict__ r,
                float* __restrict__ rTrN) {
    const int e = blockIdx.x * blockDim.x + threadIdx.x;
    const float srtr = *rTrI;
    const float spap = *pApI;
    const bool active = srtr > CG_TOL;
    const float alpha = active ? (srtr / spap) : 1.0f;
    const float rv = r[e];
    float rnew = rv;
    if (active) {
        X[e] += alpha * p[e];
        rnew = rv - alpha * Ap[e];
        r[e] = rnew;
    }
    const float s = wave_reduce_sum(rnew * rnew);
    if ((threadIdx.x & 31) == 0) atomicAdd(rTrN, s);
}

// ---------------------------------------------------------------------------
// beta = rTrNext/rTr ; p = r + beta p  (only when active)
// ---------------------------------------------------------------------------
__global__ __launch_bounds__(256)
void cg_update2(const float* __restrict__ rTrI, const float* __restrict__ rTrN,
                const float* __restrict__ r, float* __restrict__ p) {
    const int e = blockIdx.x * blockDim.x + threadIdx.x;
    const float s0 = *rTrI;
    if (s0 > CG_TOL) {
        const float beta = *rTrN / s0;
        p[e] = r[e] + beta * p[e];
    }
}

// ---------------------------------------------------------------------------
extern "C" void kernel_launch(void* const* d_in, const int* in_sizes, int n_in,
                              void* d_out, int out_size, void* d_ws, size_t ws_size,
                              hipStream_t stream) {
    const float* X   = (const float*)d_in[0];   // [B,N]
    const float* M   = (const float*)d_in[1];   // [B,N,N] SPD
    const float* RHS = (const float*)d_in[2];   // [B,N]
    float* Xout = (float*)d_out;                // [B,N]

    float* ws    = (float*)d_ws;
    float* parts = ws;                                    // KC*B*N
    float* r     = parts + (size_t)CG_KC * CG_B * CG_N;   // B*N
    float* p     = r  + CG_B * CG_N;                      // B*N
    float* Ap    = p  + CG_B * CG_N;                      // B*N
    float* rTr   = Ap + CG_B * CG_N;                      // NITER+1
    float* pAp   = rTr + (CG_NITER + 1);                  // NITER

    const dim3 mgrid(CG_N / 32, CG_KC, CG_B);             // 128 x 8 x 8 waves
    const int  eblocks = (CG_B * CG_N) / 256;             // 128

    cg_init_scalars<<<1, 64, 0, stream>>>(rTr, (CG_NITER + 1) + CG_NITER);
    cg_matvec_wmma<<<mgrid, 32, 0, stream>>>(M, X, parts);
    cg_init_rp<<<eblocks, 256, 0, stream>>>(RHS, X, parts, r, p, Xout, rTr);

    for (int i = 0; i < CG_NITER; ++i) {
        cg_matvec_wmma<<<mgrid, 32, 0, stream>>>(M, p, parts);
        cg_apdot<<<eblocks, 256, 0, stream>>>(parts, p, Ap, pAp + i);
        cg_update1<<<eblocks, 256, 0, stream>>>(rTr + i, pAp + i, p, Ap,
                                                Xout, r, rTr + i + 1);
        cg_update2<<<eblocks, 256, 0, stream>>>(rTr + i, rTr + i + 1, r, p);
    }
}